// DiSAN_82360292868116
// MI455X (gfx1250) — compile-verified
//
#include <hip/hip_runtime.h>
#include <math.h>

// Problem constants (from reference): B=4, L=200, D=100, V=30000, C=20
#define BB   4
#define LL   200
#define DD   100
#define CC   20
#define NROW (BB * LL)   // 800 rows (b,l) flattened
#define DP   112         // D padded to multiple of 16
#define D2   200         // 2*D
#define D2P  208         // 2*D padded to multiple of 16
#define CP   32          // C padded to multiple of 16
#define MS   16          // padded M for the tiny final GEMMs

typedef __attribute__((ext_vector_type(2))) float v2f;
typedef __attribute__((ext_vector_type(8))) float v8f;

// ---------------------------------------------------------------------------
// Zero-padding copy: dst (dr x dc) <- src (sr x sc), zeros outside.
// ---------------------------------------------------------------------------
__global__ void pad2d_kernel(const float* __restrict__ src, int sr, int sc,
                             float* __restrict__ dst, int dr, int dc) {
  int i = blockIdx.x * blockDim.x + threadIdx.x;
  int tot = dr * dc;
  if (i >= tot) return;
  int r = i / dc;
  int c = i % dc;
  dst[i] = (r < sr && c < sc) ? src[r * sc + c] : 0.0f;
}

__global__ void fillz_kernel(float* __restrict__ p, int n) {
  int i = blockIdx.x * blockDim.x + threadIdx.x;
  if (i < n) p[i] = 0.0f;
}

// ---------------------------------------------------------------------------
// Embedding gather into padded (NROW x DP) buffer.
// ---------------------------------------------------------------------------
__global__ void embed_kernel(const int* __restrict__ x,
                             const float* __restrict__ emb,
                             float* __restrict__ xe) {
  int i = blockIdx.x * blockDim.x + threadIdx.x;
  if (i >= NROW * DP) return;
  int row = i / DP;
  int c   = i % DP;
  xe[i] = (c < DD) ? emb[(size_t)x[row] * DD + c] : 0.0f;
}

// ---------------------------------------------------------------------------
// FP32 WMMA GEMM: C[M x N] = A[M x K] * B[K x N] (+ bias, + activation).
// One wave32 per 16x16 tile of C; V_WMMA_F32_16X16X4_F32, K stepped by 4.
// A/B/C are padded so every tile is full (EXEC all ones during WMMA).
// epi: 0=none, 1=bias+elu, 2=bias, 3=bias+relu. Bias guarded by realN.
// ---------------------------------------------------------------------------
__global__ __launch_bounds__(32) void gemm_wmma_f32(
    const float* __restrict__ A, int lda,
    const float* __restrict__ B, int ldb,
    float* __restrict__ C, int ldc,
    int K, int tilesN,
    const float* __restrict__ bias, int realN, int epi) {
  int tm   = (blockIdx.x / tilesN) * 16;
  int tn   = (blockIdx.x % tilesN) * 16;
  int lane = threadIdx.x & 31;
  int half = lane >> 4;    // 0: K pair {0,1}; 1: K pair {2,3} (per ISA layout)
  int l15  = lane & 15;

  const float* Ar = A + (size_t)(tm + l15) * lda + 2 * half; // row tm+l15
  const float* Bc = B + (size_t)(2 * half) * ldb + tn + l15; // col tn+l15

  v8f acc = {};
  for (int k = 0; k < K; k += 4) {
    v2f av, bv;
    av.x = Ar[k];                      // A[tm+l15, k + 2*half]
    av.y = Ar[k + 1];                  // A[tm+l15, k + 2*half + 1]
    bv.x = Bc[(size_t)k * ldb];        // B[k + 2*half,     tn+l15]
    bv.y = Bc[(size_t)k * ldb + ldb];  // B[k + 2*half + 1, tn+l15]
    acc = __builtin_amdgcn_wmma_f32_16x16x4_f32(
        /*neg_a=*/false, av, /*neg_b=*/false, bv,
        /*c_mod=*/(short)0, acc, /*reuse_a=*/false, /*reuse_b=*/false);
  }

  int col = tn + l15;
  float bsv = (bias != nullptr && col < realN) ? bias[col] : 0.0f;
#pragma unroll
  for (int r = 0; r < 8; ++r) {
    int row = tm + r + 8 * half;       // C/D VGPR layout
    float v = acc[r] + bsv;
    if (epi == 1)      v = (v > 0.0f) ? v : (expf(v) - 1.0f);  // elu
    else if (epi == 3) v = fmaxf(v, 0.0f);                     // relu
    C[(size_t)row * ldc + col] = v;
  }
}

// ---------------------------------------------------------------------------
// Fused attention core. One block per (b,l) row, one lane per feature e.
// att = 5*tanh((h1[b,l,e]+h2[b,m,e]+b[e])/5) is BOUNDED in [-5,5], so the
// softmax needs no running max (shift-invariant; exp(att) <= e^5, sum fits
// easily in f32). Masked entries get weight exactly 0 — identical to the
// reference, where exp(-1e13 - max) underflows to 0 whenever any unmasked
// entry exists. The only special case is "all entries masked": reference
// softmax of all-equal -INF is uniform => s = mean_m(h); we track sum(h)
// and select it when the denominator is 0.
// exp(att) is computed ONCE and shared by both directions (fw/bw differ
// only by the causal kill). Fully branchless: 1 tanh + 1 exp + ~8 FMAs
// per (m,e). Operands (h/h1/h2 ~350 KB each) are L2/WGP$-resident.
// ---------------------------------------------------------------------------
__global__ __launch_bounds__(128) void attend_kernel(
    const float* __restrict__ h,    // NROW x DP
    const float* __restrict__ h1,   // NROW x DP
    const float* __restrict__ h2,   // NROW x DP
    const float* __restrict__ bvec, // DD
    const unsigned char* __restrict__ mask, // B*L bools
    float* __restrict__ s_fw,       // NROW x DP (pads zeroed)
    float* __restrict__ s_bw) {
  int row = blockIdx.x;             // b*LL + l
  int b = row / LL;
  int l = row % LL;
  int e = threadIdx.x;
  if (e >= DP) return;
  if (e >= DD) {                    // keep padded columns exactly zero
    s_fw[(size_t)row * DP + e] = 0.0f;
    s_bw[(size_t)row * DP + e] = 0.0f;
    return;
  }

  float h1v = h1[(size_t)row * DP + e] + bvec[e];
  bool ml = (mask[b * LL + l] != 0);

  float dF = 0.0f, nF = 0.0f;       // fw denominator / numerator
  float dB = 0.0f, nB = 0.0f;       // bw denominator / numerator
  float hsum = 0.0f;                // for the all-masked uniform fallback

  for (int m = 0; m < LL; ++m) {
    int r2 = b * LL + m;
    float att = 5.0f * tanhf((h1v + h2[(size_t)r2 * DP + e]) * 0.2f);
    float hv  = h[(size_t)r2 * DP + e];
    bool dead = (mask[r2] != 0) && !ml;        // m2d mask kills both dirs

    float ev = expf(att);                      // shared by fw and bw
    float wF = (dead || (m <= l)) ? 0.0f : ev; // fw_fill: m <= l -> -INF
    float wB = (dead || (m >= l)) ? 0.0f : ev; // bw_fill: m >= l -> -INF

    dF += wF;  nF = fmaf(wF, hv, nF);
    dB += wB;  nB = fmaf(wB, hv, nB);
    hsum += hv;
  }

  float uni = hsum * (1.0f / (float)LL);       // softmax(all -INF) = uniform
  s_fw[(size_t)row * DP + e] = (dF > 0.0f) ? (nF / dF) : uni;
  s_bw[(size_t)row * DP + e] = (dB > 0.0f) ? (nB / dB) : uni;
}

// ---------------------------------------------------------------------------
// Gate + combine: f = sigmoid(g1 + g2 + Wf2_b); u[:, off+e] = f*h + (1-f)*s
// ---------------------------------------------------------------------------
__global__ void combine_kernel(const float* __restrict__ g1,
                               const float* __restrict__ g2,
                               const float* __restrict__ Wf2_b,
                               const float* __restrict__ h,
                               const float* __restrict__ s,
                               float* __restrict__ u, int coloff) {
  int i = blockIdx.x * blockDim.x + threadIdx.x;
  if (i >= NROW * DD) return;
  int row = i / DD;
  int e   = i % DD;
  size_t p = (size_t)row * DP + e;
  float xg = g1[p] + g2[p] + Wf2_b[e];
  float f = 1.0f / (1.0f + expf(-xg));
  u[(size_t)row * D2P + coloff + e] = f * h[p] + (1.0f - f) * s[p];
}

// ---------------------------------------------------------------------------
// s_s[b, j] = sum_l u[b,l,j] * att_s[b,l,j], into padded (MS x D2P) buffer.
// Rows b>=BB and pad cols come out zero (att_s pads are exactly 0).
// ---------------------------------------------------------------------------
__global__ void reduce_ss_kernel(const float* __restrict__ u,
                                 const float* __restrict__ as,
                                 float* __restrict__ ss) {
  int i = blockIdx.x * blockDim.x + threadIdx.x;
  if (i >= MS * D2P) return;
  int bq = i / D2P;
  int j  = i % D2P;
  float acc = 0.0f;
  if (bq < BB) {
    for (int l = 0; l < LL; ++l) {
      size_t r = (size_t)(bq * LL + l) * D2P;
      acc = fmaf(u[r + j], as[r + j], acc);
    }
  }
  ss[i] = acc;
}

__global__ void extract_kernel(const float* __restrict__ op,
                               float* __restrict__ out) {
  int i = blockIdx.x * blockDim.x + threadIdx.x;
  if (i >= BB * CC) return;
  int b = i / CC;
  int c = i % CC;
  out[i] = op[b * CP + c];
}

// ---------------------------------------------------------------------------
// Host-side orchestration
// ---------------------------------------------------------------------------
static inline int ceil_div(int a, int b) { return (a + b - 1) / b; }

extern "C" void kernel_launch(void* const* d_in, const int* in_sizes, int n_in,
                              void* d_out, int out_size, void* d_ws, size_t ws_size,
                              hipStream_t stream) {
  (void)in_sizes; (void)n_in; (void)out_size; (void)ws_size;

  const int*           x     = (const int*)d_in[0];
  /* d_in[1] = lengths (unused: all == L) */
  const unsigned char* mask  = (const unsigned char*)d_in[2];
  const float* emb   = (const float*)d_in[3];
  const float* Wh_w  = (const float*)d_in[4];
  const float* Wh_b  = (const float*)d_in[5];
  const float* W1_w  = (const float*)d_in[6];
  const float* W2_w  = (const float*)d_in[7];
  const float* bvec  = (const float*)d_in[8];
  const float* Wf1_w = (const float*)d_in[9];
  const float* Wf2_w = (const float*)d_in[10];
  const float* Wf2_b = (const float*)d_in[11];
  const float* Ws1_w = (const float*)d_in[12];
  const float* Ws1_b = (const float*)d_in[13];
  const float* Ws_w  = (const float*)d_in[14];
  const float* Ws_b  = (const float*)d_in[15];
  const float* F1_w  = (const float*)d_in[16];
  const float* F1_b  = (const float*)d_in[17];
  const float* F2_w  = (const float*)d_in[18];
  const float* F2_b  = (const float*)d_in[19];
  float* out = (float*)d_out;

  // Carve workspace (floats)
  float* ws = (float*)d_ws;
  size_t o = 0;
  float* xe   = ws + o; o += (size_t)NROW * DP;
  float* h    = ws + o; o += (size_t)NROW * DP;
  float* h1   = ws + o; o += (size_t)NROW * DP;
  float* h2   = ws + o; o += (size_t)NROW * DP;
  float* sfw  = ws + o; o += (size_t)NROW * DP;
  float* sbw  = ws + o; o += (size_t)NROW * DP;
  float* g1   = ws + o; o += (size_t)NROW * DP;
  float* g2   = ws + o; o += (size_t)NROW * DP;
  float* u    = ws + o; o += (size_t)NROW * D2P;
  float* t    = ws + o; o += (size_t)NROW * D2P;
  float* as   = ws + o; o += (size_t)NROW * D2P;
  float* ssb  = ws + o; o += (size_t)MS * D2P;
  float* z    = ws + o; o += (size_t)MS * DP;
  float* op   = ws + o; o += (size_t)MS * CP;
  float* Wh_p  = ws + o; o += (size_t)DP * DP;
  float* W1_p  = ws + o; o += (size_t)DP * DP;
  float* W2_p  = ws + o; o += (size_t)DP * DP;
  float* Wf1_p = ws + o; o += (size_t)DP * DP;
  float* Wf2_p = ws + o; o += (size_t)DP * DP;
  float* Ws1_p = ws + o; o += (size_t)D2P * D2P;
  float* Ws_p  = ws + o; o += (size_t)D2P * D2P;
  float* F1_p  = ws + o; o += (size_t)D2P * DP;
  float* F2_p  = ws + o; o += (size_t)DP * CP;

  const int TPB = 256;
  auto padw = [&](const float* src, int sr, int sc, float* dst, int dr, int dc) {
    pad2d_kernel<<<ceil_div(dr * dc, TPB), TPB, 0, stream>>>(src, sr, sc, dst, dr, dc);
  };

  // 1) Pad weights into WMMA-friendly buffers
  padw(Wh_w,  DD, DD, Wh_p,  DP,  DP);
  padw(W1_w,  DD, DD, W1_p,  DP,  DP);
  padw(W2_w,  DD, DD, W2_p,  DP,  DP);
  padw(Wf1_w, DD, DD, Wf1_p, DP,  DP);
  padw(Wf2_w, DD, DD, Wf2_p, DP,  DP);
  padw(Ws1_w, D2, D2, Ws1_p, D2P, D2P);
  padw(Ws_w,  D2, D2, Ws_p,  D2P, D2P);
  padw(F1_w,  D2, DD, F1_p,  D2P, DP);
  padw(F2_w,  DD, CC, F2_p,  DP,  CP);

  // 2) Embedding gather
  embed_kernel<<<ceil_div(NROW * DP, TPB), TPB, 0, stream>>>(x, emb, xe);

  auto gemm = [&](const float* A, int lda, const float* B, int ldb,
                  float* C, int ldc, int M, int N, int K,
                  const float* bias, int realN, int epi) {
    int tilesN = N / 16;
    int ntiles = (M / 16) * tilesN;
    gemm_wmma_f32<<<ntiles, 32, 0, stream>>>(A, lda, B, ldb, C, ldc,
                                             K, tilesN, bias, realN, epi);
  };

  // 3) h = elu(xe @ Wh + Wh_b); h1 = h @ W1; h2 = h @ W2
  gemm(xe, DP, Wh_p, DP, h,  DP, NROW, DP, DD, Wh_b, DD, /*elu*/1);
  gemm(h,  DP, W1_p, DP, h1, DP, NROW, DP, DD, nullptr, 0, 0);
  gemm(h,  DP, W2_p, DP, h2, DP, NROW, DP, DD, nullptr, 0, 0);

  // 4) Fused masked bidirectional attention (branchless shared-exp softmax)
  attend_kernel<<<NROW, 128, 0, stream>>>(h, h1, h2, bvec, mask, sfw, sbw);

  // 5) Gates: g2 = h @ Wf2 (shared); per direction g1 = s @ Wf1, then combine
  fillz_kernel<<<ceil_div(NROW * D2P, TPB), TPB, 0, stream>>>(u, NROW * D2P);
  gemm(h, DP, Wf2_p, DP, g2, DP, NROW, DP, DD, nullptr, 0, 0);
  gemm(sfw, DP, Wf1_p, DP, g1, DP, NROW, DP, DD, nullptr, 0, 0);
  combine_kernel<<<ceil_div(NROW * DD, TPB), TPB, 0, stream>>>(g1, g2, Wf2_b, h, sfw, u, 0);
  gemm(sbw, DP, Wf1_p, DP, g1, DP, NROW, DP, DD, nullptr, 0, 0);
  combine_kernel<<<ceil_div(NROW * DD, TPB), TPB, 0, stream>>>(g1, g2, Wf2_b, h, sbw, u, DD);

  // 6) att_s = elu(u @ Ws1 + b1) @ Ws + b   (K = 200)
  gemm(u, D2P, Ws1_p, D2P, t,  D2P, NROW, D2P, D2, Ws1_b, D2, /*elu*/1);
  gemm(t, D2P, Ws_p,  D2P, as, D2P, NROW, D2P, D2, Ws_b,  D2, /*bias*/2);

  // 7) s_s = sum_l u * att_s
  reduce_ss_kernel<<<ceil_div(MS * D2P, TPB), TPB, 0, stream>>>(u, as, ssb);

  // 8) out = relu(s_s @ F1 + b) @ F2 + b
  gemm(ssb, D2P, F1_p, DP, z,  DP, MS, DP, D2, F1_b, DD, /*relu*/3);
  gemm(z,   DP,  F2_p, CP, op, CP, MS, CP, DD, F2_b, CC, /*bias*/2);

  // 9) Extract the real (B x C) block
  extract_kernel<<<1, 128, 0, stream>>>(op, out);
}